// PR_N_72353019068520
// MI455X (gfx1250) — compile-verified
//
#include <hip/hip_runtime.h>
#include <math.h>

// Problem constants (from the reference)
#define KC     5                // classes
#define NSUP   5                // support per class
#define DIM    640
#define QN     40000
#define NSAMP  50
#define NTILES (QN / 16)        // 2500 query tiles of 16
#define PSTR   648              // LDS proto row stride (648 % 64 = 8 -> conflict-free)
#define NREP   8                // replicated global accumulators (atomic contention)

typedef __attribute__((ext_vector_type(2))) float v2f;
typedef __attribute__((ext_vector_type(8))) float v8f;

static __device__ __forceinline__ v8f wmma_f32(v2f a, v2f b, v8f c) {
  // D = A(16x4 f32) * B(4x16 f32) + C(16x16 f32)
  return __builtin_amdgcn_wmma_f32_16x16x4_f32(false, a, false, b, (short)0, c, false, false);
}

// ---------------------------------------------------------------------------
// Kernel 1: per-class support sums / sqsums, proto_e, |proto_e|^2, noise sums.
// grid = KC blocks, 128 threads.
// ---------------------------------------------------------------------------
__global__ void prep_support(const float* __restrict__ feat,
                             const float* __restrict__ noise,
                             float* __restrict__ ssum, float* __restrict__ ssq,
                             float* __restrict__ protoE, float* __restrict__ nsum,
                             float* __restrict__ psqE) {
  const int c = blockIdx.x;
  float psqPart = 0.f;
  for (int d = threadIdx.x; d < DIM; d += blockDim.x) {
    float s = 0.f, q2 = 0.f;
    #pragma unroll
    for (int n = 0; n < NSUP; ++n) {
      float v = feat[(c * NSUP + n) * DIM + d];
      s += v; q2 += v * v;
    }
    float ns = 0.f;
    for (int s2 = 0; s2 < NSAMP; ++s2)
      ns += noise[(c * NSAMP + s2) * DIM + d];
    ssum[c * DIM + d] = s;
    ssq[c * DIM + d]  = q2;
    float p = s * (1.0f / NSUP);
    protoE[c * DIM + d] = p;
    nsum[c * DIM + d]   = ns;
    psqPart += p * p;
  }
  atomicAdd(&psqE[c], psqPart);
}

// ---------------------------------------------------------------------------
// Kernel 2: WMMA pass 1 over queries: dist_e -> pred, masked class sums,
// counts, loss accumulation. One 16-query tile per wave.
// ---------------------------------------------------------------------------
__global__ void pass_e(const float* __restrict__ feat,
                       const int*   __restrict__ label1,   // label[1], int32
                       const float* __restrict__ protoE,
                       const float* __restrict__ psqEg,
                       float* __restrict__ qsumR,          // NREP x KC*DIM
                       float* __restrict__ qsqR,           // NREP x KC*DIM
                       float* __restrict__ cntG,           // KC
                       float* __restrict__ lossG) {
  __shared__ float sProto[(KC + 1) * PSTR];   // row KC is all-zero padding
  __shared__ float sTile[8][16 * 17];
  __shared__ int   sPred[8][16];
  __shared__ float sAccS[KC * DIM];
  __shared__ float sAccQ[KC * DIM];
  __shared__ float sCnt[8];
  __shared__ float sLoss;

  const int tid = threadIdx.x;
  for (int i = tid; i < KC * DIM; i += blockDim.x) {
    sProto[(i / DIM) * PSTR + (i % DIM)] = protoE[i];
    sAccS[i] = 0.f; sAccQ[i] = 0.f;
  }
  for (int i = tid; i < DIM; i += blockDim.x) sProto[KC * PSTR + i] = 0.f;
  if (tid < 8) sCnt[tid] = 0.f;
  if (tid == 0) sLoss = 0.f;
  __syncthreads();

  float psq[KC];
  #pragma unroll
  for (int c = 0; c < KC; ++c) psq[c] = psqEg[c];

  const float* queries = feat + KC * NSUP * DIM;
  const int lane = tid & 31;
  const int wave = tid >> 5;
  const int h    = lane >> 4;        // K-half: 0 -> K=0,1 ; 1 -> K=2,3
  const int nl   = lane & 15;        // A row within tile == B column (class)
  const int cn   = (nl < KC) ? nl : KC;  // classes >= KC read the zero row
  const int koff = 2 * h;
  const int wavesTotal = gridDim.x * (blockDim.x >> 5);
  const int wgid = blockIdx.x * (blockDim.x >> 5) + wave;

  float* myTile = &sTile[wave][0];
  int*   myPred = &sPred[wave][0];

  for (int t = wgid; t < NTILES; t += wavesTotal) {
    const int rowBase = t * 16;
    const float* qrow = queries + (size_t)(rowBase + nl) * DIM + koff;
    const float* brow = sProto + cn * PSTR + koff;

    v8f acc0 = {0.f,0.f,0.f,0.f,0.f,0.f,0.f,0.f};
    v8f acc1 = {0.f,0.f,0.f,0.f,0.f,0.f,0.f,0.f};
    float sq = 0.f;
    #pragma unroll 4
    for (int db = 0; db < DIM; db += 8) {
      v2f a0 = *(const v2f*)(qrow + db);
      v2f b0 = *(const v2f*)(brow + db);
      v2f a1 = *(const v2f*)(qrow + db + 4);
      v2f b1 = *(const v2f*)(brow + db + 4);
      sq += a0.x * a0.x + a0.y * a0.y + a1.x * a1.x + a1.y * a1.y;
      acc0 = wmma_f32(a0, b0, acc0);
      acc1 = wmma_f32(a1, b1, acc1);
    }
    v8f acc = acc0 + acc1;
    sq += __shfl_xor(sq, 16);   // combine K-halves -> full |q_row|^2 per lane

    #pragma unroll
    for (int v = 0; v < 8; ++v) myTile[(v + 8 * h) * 17 + nl] = acc[v];
    asm volatile("s_wait_dscnt 0" ::: "memory");

    if (h == 0) {   // lanes 0..15: one query row each
      float de[KC];
      #pragma unroll
      for (int c = 0; c < KC; ++c) {
        float dot = myTile[nl * 17 + c];
        float d2  = sq + psq[c] - 2.f * dot;
        de[c] = sqrtf(fmaxf(d2, 1e-12f));
      }
      int pred = 0; float best = de[0];
      #pragma unroll
      for (int c = 1; c < KC; ++c) if (de[c] < best) { best = de[c]; pred = c; }
      myPred[nl] = pred;
      atomicAdd(&sCnt[pred], 1.0f);

      // loss: double log-softmax of x = -de, gathered at label
      float m1 = -de[0];
      #pragma unroll
      for (int c = 1; c < KC; ++c) m1 = fmaxf(m1, -de[c]);
      float s1 = 0.f;
      #pragma unroll
      for (int c = 0; c < KC; ++c) s1 += expf(-de[c] - m1);
      float ls1 = logf(s1);
      float l1[KC];
      #pragma unroll
      for (int c = 0; c < KC; ++c) l1[c] = -de[c] - m1 - ls1;
      float m2 = l1[0];
      #pragma unroll
      for (int c = 1; c < KC; ++c) m2 = fmaxf(m2, l1[c]);
      float s2 = 0.f;
      #pragma unroll
      for (int c = 0; c < KC; ++c) s2 += expf(l1[c] - m2);
      float ls2 = logf(s2);
      int lbl = label1[rowBase + nl];
      atomicAdd(&sLoss, l1[lbl] - m2 - ls2);
    }
    asm volatile("s_wait_dscnt 0" ::: "memory");

    // masked per-class accumulation (re-reads tile; hits L2/L0)
    for (int r = 0; r < 16; ++r) {
      const int cr = myPred[r];
      const float* qr = queries + (size_t)(rowBase + r) * DIM;
      float* as = sAccS + cr * DIM;
      float* aq = sAccQ + cr * DIM;
      for (int d = lane; d < DIM; d += 32) {
        float v = qr[d];
        atomicAdd(&as[d], v);
        atomicAdd(&aq[d], v * v);
      }
    }
  }
  __syncthreads();

  float* qs = qsumR + (blockIdx.x % NREP) * (KC * DIM);
  float* q2 = qsqR  + (blockIdx.x % NREP) * (KC * DIM);
  for (int i = tid; i < KC * DIM; i += blockDim.x) {
    atomicAdd(&qs[i], sAccS[i]);
    atomicAdd(&q2[i], sAccQ[i]);
  }
  if (tid < KC) atomicAdd(&cntG[tid], sCnt[tid]);
  if (tid == 0) atomicAdd(lossG, sLoss);
}

// ---------------------------------------------------------------------------
// Kernel 3: proto_new, |proto_new|^2, final loss. grid = KC blocks.
// ---------------------------------------------------------------------------
__global__ void make_proto_new(const float* __restrict__ ssum, const float* __restrict__ ssq,
                               const float* __restrict__ nsum,
                               const float* __restrict__ qsumR, const float* __restrict__ qsqR,
                               const float* __restrict__ cntG, const float* __restrict__ lossG,
                               float* __restrict__ protoN, float* __restrict__ psqN,
                               float* __restrict__ dout) {
  const int c = blockIdx.x;
  const float cN = (float)NSUP + cntG[c];
  float part = 0.f;
  for (int d = threadIdx.x; d < DIM; d += blockDim.x) {
    const int i = c * DIM + d;
    float qs = 0.f, q2 = 0.f;
    #pragma unroll
    for (int r = 0; r < NREP; ++r) {
      qs += qsumR[r * (KC * DIM) + i];
      q2 += qsqR [r * (KC * DIM) + i];
    }
    float mean = (ssum[i] + qs) / cN;
    float sqs  = ssq[i] + q2;
    float var  = (sqs - cN * mean * mean) / (cN - 1.0f);
    float stdc = sqrtf(fmaxf(var, 0.0f));
    float pn   = (ssum[i] + (float)NSAMP * mean + stdc * nsum[i]) / (float)(NSUP + NSAMP);
    protoN[i] = pn;
    part += pn * pn;
  }
  atomicAdd(&psqN[c], part);
  if (c == 0 && threadIdx.x == 0) dout[QN * KC] = -lossG[0] / (float)QN;
}

// ---------------------------------------------------------------------------
// Kernel 4: WMMA pass 2: dist_new -> softmax -> y_pred.
// ---------------------------------------------------------------------------
__global__ void pass_new(const float* __restrict__ feat,
                         const float* __restrict__ protoN,
                         const float* __restrict__ psqNg,
                         float* __restrict__ yout) {
  __shared__ float sProto[(KC + 1) * PSTR];
  __shared__ float sTile[8][16 * 17];

  const int tid = threadIdx.x;
  for (int i = tid; i < KC * DIM; i += blockDim.x)
    sProto[(i / DIM) * PSTR + (i % DIM)] = protoN[i];
  for (int i = tid; i < DIM; i += blockDim.x) sProto[KC * PSTR + i] = 0.f;
  __syncthreads();

  float psq[KC];
  #pragma unroll
  for (int c = 0; c < KC; ++c) psq[c] = psqNg[c];

  const float* queries = feat + KC * NSUP * DIM;
  const int lane = tid & 31;
  const int wave = tid >> 5;
  const int h    = lane >> 4;
  const int nl   = lane & 15;
  const int cn   = (nl < KC) ? nl : KC;
  const int koff = 2 * h;
  const int wavesTotal = gridDim.x * (blockDim.x >> 5);
  const int wgid = blockIdx.x * (blockDim.x >> 5) + wave;
  float* myTile = &sTile[wave][0];

  for (int t = wgid; t < NTILES; t += wavesTotal) {
    const int rowBase = t * 16;
    const float* qrow = queries + (size_t)(rowBase + nl) * DIM + koff;
    const float* brow = sProto + cn * PSTR + koff;

    v8f acc0 = {0.f,0.f,0.f,0.f,0.f,0.f,0.f,0.f};
    v8f acc1 = {0.f,0.f,0.f,0.f,0.f,0.f,0.f,0.f};
    float sq = 0.f;
    #pragma unroll 4
    for (int db = 0; db < DIM; db += 8) {
      v2f a0 = *(const v2f*)(qrow + db);
      v2f b0 = *(const v2f*)(brow + db);
      v2f a1 = *(const v2f*)(qrow + db + 4);
      v2f b1 = *(const v2f*)(brow + db + 4);
      sq += a0.x * a0.x + a0.y * a0.y + a1.x * a1.x + a1.y * a1.y;
      acc0 = wmma_f32(a0, b0, acc0);
      acc1 = wmma_f32(a1, b1, acc1);
    }
    v8f acc = acc0 + acc1;
    sq += __shfl_xor(sq, 16);

    #pragma unroll
    for (int v = 0; v < 8; ++v) myTile[(v + 8 * h) * 17 + nl] = acc[v];
    asm volatile("s_wait_dscnt 0" ::: "memory");

    if (h == 0) {
      float dn[KC];
      #pragma unroll
      for (int c = 0; c < KC; ++c) {
        float dot = myTile[nl * 17 + c];
        float d2  = sq + psq[c] - 2.f * dot;
        dn[c] = sqrtf(fmaxf(d2, 1e-12f));
      }
      float m = -dn[0];
      #pragma unroll
      for (int c = 1; c < KC; ++c) m = fmaxf(m, -dn[c]);
      float e[KC]; float s = 0.f;
      #pragma unroll
      for (int c = 0; c < KC; ++c) { e[c] = expf(-dn[c] - m); s += e[c]; }
      float inv = 1.0f / s;
      float* yo = yout + (size_t)(rowBase + nl) * KC;
      #pragma unroll
      for (int c = 0; c < KC; ++c) yo[c] = e[c] * inv;
    }
    asm volatile("s_wait_dscnt 0" ::: "memory");  // protect sTile before next tile
  }
}

// ---------------------------------------------------------------------------
// Workspace layout (floats):
//  0      ssum   [KC*DIM]      3200
//  3200   ssq    [KC*DIM]      3200
//  6400   protoE [KC*DIM]      3200
//  9600   nsum   [KC*DIM]      3200
//  12800  protoN [KC*DIM]      3200
//  16000  qsumR  [NREP*KC*DIM] 25600
//  41600  qsqR   [NREP*KC*DIM] 25600
//  67200  cnt    [8]
//  67208  loss   [4 pad]
//  67212  psqE   [8]
//  67220  psqN   [8]
//  total  67228 floats (~263 KB)
// ---------------------------------------------------------------------------
extern "C" void kernel_launch(void* const* d_in, const int* in_sizes, int n_in,
                              void* d_out, int out_size, void* d_ws, size_t ws_size,
                              hipStream_t stream) {
  const float* feat  = (const float*)d_in[0];
  const int*   label = (const int*)d_in[1];     // (2, Q) int32 (JAX x64 off)
  const float* noise = (const float*)d_in[2];
  float* out = (float*)d_out;
  float* ws  = (float*)d_ws;

  float* ssum   = ws;
  float* ssq    = ws + 3200;
  float* protoE = ws + 6400;
  float* nsum   = ws + 9600;
  float* protoN = ws + 12800;
  float* qsumR  = ws + 16000;
  float* qsqR   = ws + 41600;
  float* cntG   = ws + 67200;
  float* lossG  = ws + 67208;
  float* psqE   = ws + 67212;
  float* psqN   = ws + 67220;

  hipMemsetAsync(d_ws, 0, (size_t)67228 * sizeof(float), stream);

  prep_support<<<KC, 128, 0, stream>>>(feat, noise, ssum, ssq, protoE, nsum, psqE);

  // 312 blocks * 8 waves = 2496 waves for 2500 tiles (near-perfect balance)
  pass_e<<<312, 256, 0, stream>>>(feat, label + QN, protoE, psqE,
                                  qsumR, qsqR, cntG, lossG);

  make_proto_new<<<KC, 128, 0, stream>>>(ssum, ssq, nsum, qsumR, qsqR,
                                         cntG, lossG, protoN, psqN, out);

  pass_new<<<312, 256, 0, stream>>>(feat, protoN, psqN, out);
}